// PCAGeometry_39324720562658
// MI455X (gfx1250) — compile-verified
//
#include <hip/hip_runtime.h>
#include <math.h>

typedef __attribute__((ext_vector_type(2))) float v2f;
typedef __attribute__((ext_vector_type(8))) float v8f;

#define KNBR 64            // neighbors per point
#define PTS_PER_WAVE 5     // 5 points * 3 coords = 15 rows + 1 ones-row = 16 (WMMA M)
#define WAVES_PER_BLOCK 8  // 256 threads, wave32
#define PTS_PER_BLOCK (PTS_PER_WAVE * WAVES_PER_BLOCK)  // 40
#define LDS_PT_STRIDE 200  // 192 floats payload + 8 pad (bank spread)

__global__ __launch_bounds__(256)
void pca_geometry_kernel(const float* __restrict__ xyz,  // (P, 64, 3)
                         float* __restrict__ out,        // (P, 9)
                         int P)
{
    __shared__ __align__(16) float s_pts[WAVES_PER_BLOCK][PTS_PER_WAVE * LDS_PT_STRIDE];
    __shared__ float s_ones[192];                     // ones-row source (row 15 of A)
    __shared__ float s_d[WAVES_PER_BLOCK][16 * 17];   // 16x16 D matrix, padded rows

    const int wave = threadIdx.x >> 5;
    const int lane = threadIdx.x & 31;
    const int hi   = (lane >> 4) & 1;   // which K-half this lane holds
    const int m    = lane & 15;         // WMMA row index (M)

    const int p0 = blockIdx.x * PTS_PER_BLOCK + wave * PTS_PER_WAVE;

    if (threadIdx.x < 192) s_ones[threadIdx.x] = 1.0f;

    // ---- Stage 5 points: coalesced float2 global loads -> LDS (3 uniform iters/lane) ----
    #pragma unroll
    for (int g = 0; g < PTS_PER_WAVE; ++g) {
        int pidx = p0 + g;
        size_t pp = (size_t)((pidx < P) ? pidx : (P - 1));   // clamp tail (store-guarded later)
        const float2* src = (const float2*)(xyz + pp * (size_t)(KNBR * 3));
        float2* dst = (float2*)&s_pts[wave][g * LDS_PT_STRIDE];
        #pragma unroll
        for (int it = 0; it < 3; ++it)                       // 96 float2 per point, 32 lanes
            dst[lane + it * 32] = src[lane + it * 32];
    }
    __syncthreads();

    // ---- Gather all A fragments to registers (unconditional ds loads) ----
    // Row m<15: point g = m/3, coord c = m%3. Row 15 reads the LDS ones block.
    const bool onesRow = (m == 15);
    const int  gq = m / 3;                   // 5 for m==15 (unused then)
    const int  cq = m - 3 * gq;
    const float* rp = onesRow ? s_ones
                              : &s_pts[wave][gq * LDS_PT_STRIDE + cq];  // element k at rp[3*k]

    v2f frag[16];
    #pragma unroll
    for (int ck = 0; ck < 16; ++ck) {
        const int k0 = ck * 4 + 2 * hi;      // K base held by this lane for chunk ck
        frag[ck].x = rp[3 * k0];
        frag[ck].y = rp[3 * k0 + 3];
    }

    // ---- D = X * X^T via 16x V_WMMA_F32_16X16X4_F32, two interleaved accumulators ----
    // B = A^T has an identical per-lane register image for this shape -> pass frag as both.
    v8f acc0 = {0.f,0.f,0.f,0.f,0.f,0.f,0.f,0.f};
    v8f acc1 = {0.f,0.f,0.f,0.f,0.f,0.f,0.f,0.f};
    #pragma unroll
    for (int ck = 0; ck < 16; ck += 2) {
        acc0 = __builtin_amdgcn_wmma_f32_16x16x4_f32(false, frag[ck],   false, frag[ck],
                                                     (short)0, acc0, false, false);
        acc1 = __builtin_amdgcn_wmma_f32_16x16x4_f32(false, frag[ck+1], false, frag[ck+1],
                                                     (short)0, acc1, false, false);
    }

    // ---- Spill D (16x16) to LDS: lane holds D[r + 8*hi][m] in acc[r] ----
    #pragma unroll
    for (int r = 0; r < 8; ++r)
        s_d[wave][(r + 8 * hi) * 17 + m] = acc0[r] + acc1[r];
    __syncthreads();

    // ---- 5 lanes per wave: covariance -> analytic 3x3 eigensolve -> features ----
    if (lane < PTS_PER_WAVE) {
        const float* D = s_d[wave];
        const int b = 3 * lane;

        const float s0 = D[15 * 17 + b + 0];   // per-row sums from the ones-row
        const float s1 = D[15 * 17 + b + 1];
        const float s2 = D[15 * 17 + b + 2];
        const float invK  = 1.0f / (float)KNBR;
        const float denom = 1.0f / ((float)(KNBR - 1) + 1e-8f);

        const float a00 = (D[(b+0)*17 + b+0] - s0*s0*invK) * denom;
        const float a01 = (D[(b+0)*17 + b+1] - s0*s1*invK) * denom;
        const float a02 = (D[(b+0)*17 + b+2] - s0*s2*invK) * denom;
        const float a11 = (D[(b+1)*17 + b+1] - s1*s1*invK) * denom;
        const float a12 = (D[(b+1)*17 + b+2] - s1*s2*invK) * denom;
        const float a22 = (D[(b+2)*17 + b+2] - s2*s2*invK) * denom;

        // Trigonometric eigenvalues of symmetric 3x3
        const float q  = (a00 + a11 + a22) * (1.0f / 3.0f);
        const float p1 = a01*a01 + a02*a02 + a12*a12;
        const float b00 = a00 - q, b11 = a11 - q, b22 = a22 - q;
        const float p2 = b00*b00 + b11*b11 + b22*b22 + 2.0f * p1;

        float l1, l2, l3;
        if (p2 <= 1e-30f) {
            l1 = l2 = l3 = q;
        } else {
            const float pr   = sqrtf(p2 * (1.0f / 6.0f));
            const float invp = 1.0f / pr;
            const float c00 = b00*invp, c11 = b11*invp, c22 = b22*invp;
            const float c01 = a01*invp, c02 = a02*invp, c12 = a12*invp;
            float r = 0.5f * (c00*(c11*c22 - c12*c12)
                            - c01*(c01*c22 - c12*c02)
                            + c02*(c01*c12 - c11*c02));
            r = fminf(1.0f, fmaxf(-1.0f, r));
            const float phi = acosf(r) * (1.0f / 3.0f);
            l1 = q + 2.0f * pr * cosf(phi);                       // largest
            l3 = q + 2.0f * pr * cosf(phi + 2.0943951023931953f); // smallest
            l2 = 3.0f * q - l1 - l3;
        }

        // Eigenvector of largest eigenvalue: max-norm cross product of rows of (A - l1*I)
        const float m00 = a00 - l1, m11 = a11 - l1, m22 = a22 - l1;
        const float c01x = a01*a12 - a02*m11, c01y = a02*a01 - m00*a12, c01z = m00*m11 - a01*a01;
        const float c02x = a01*m22 - a02*a12, c02y = a02*a02 - m00*m22, c02z = m00*a12 - a01*a02;
        const float c12x = m11*m22 - a12*a12, c12y = a12*a02 - a01*m22, c12z = a01*a12 - m11*a02;
        const float n01 = c01x*c01x + c01y*c01y + c01z*c01z;
        const float n02 = c02x*c02x + c02y*c02y + c02z*c02z;
        const float n12 = c12x*c12x + c12y*c12y + c12z*c12z;
        float vz, nn;
        if (n01 >= n02 && n01 >= n12)      { vz = c01z; nn = n01; }
        else if (n02 >= n12)               { vz = c02z; nn = n02; }
        else                               { vz = c12z; nn = n12; }
        const float ez = fabsf(vz) * rsqrtf(fmaxf(nn, 1e-30f));

        // Features (clip eigenvalues like the reference)
        const float l1c = fmaxf(l1, 1e-10f);
        const float l2c = fmaxf(l2, 1e-10f);
        const float l3c = fmaxf(l3, 1e-10f);
        const float inv_l1 = 1.0f / l1c;

        const int p = p0 + lane;
        if (p < P) {
            float* o = out + (size_t)p * 9;
            o[0] = (l1c - l2c) * inv_l1;                     // linearity
            o[1] = (l2c - l3c) * inv_l1;                     // planarity
            o[2] = l3c * inv_l1;                             // scattering
            o[3] = cbrtf(fmaxf(l1c * l2c * l3c, 0.0f));      // omnivariance
            o[4] = (l1c - l3c) * inv_l1;                     // anisotropy
            o[5] = l1c;
            o[6] = l2c;
            o[7] = l3c;
            o[8] = ez;                                       // |z| of principal eigenvector
        }
    }
}

extern "C" void kernel_launch(void* const* d_in, const int* in_sizes, int n_in,
                              void* d_out, int out_size, void* d_ws, size_t ws_size,
                              hipStream_t stream) {
    const float* xyz = (const float*)d_in[0];
    float* out = (float*)d_out;
    const int P = in_sizes[0] / (KNBR * 3);        // B*N points
    const int blocks = (P + PTS_PER_BLOCK - 1) / PTS_PER_BLOCK;
    pca_geometry_kernel<<<blocks, 256, 0, stream>>>(xyz, out, P);
}